// PermutationInvariantMetric_7876970020940
// MI455X (gfx1250) — compile-verified
//
#include <hip/hip_runtime.h>

// Permutation-invariant metric, B=16, C=4, T=256000 (derived at launch).
//
// cost[b,i,j]*T = Sp_i + St_j - 2*X_ij  with  X_ij = sum_t p_i t_j,
// Sp_i = sum_t p_i^2, St_j = sum_t t_j^2.
// All three come from one packed V_WMMA_F32_16X16X4_F32 accumulation:
//   A rows [8a + r]: r<4 -> pred ch r, r>=4 -> tgt ch r-4, K-chunk a
//   B cols [8a + s]: s<4 -> tgt  ch s, s>=4 -> pred ch s-4, K-chunk a
// Diagonal 8x8 blocks of D carry p_i*t_j (X), p_i*p_i (Sp, col 4+i),
// t_j*t_j (St, row 4+j). HBM-bound: 128 MB streamed once ~ 5.5 us @ 23.3 TB/s.
// Two independent accumulator chains + software pipelining keep 4 b64 loads
// in flight over each WMMA pair; all loop control is scalar (readfirstlane).

typedef __attribute__((ext_vector_type(2))) float v2f;
typedef __attribute__((ext_vector_type(8))) float v8f;

#define NSLICE 32   // t-slices per batch (blocks in x)
#define WPB    8    // waves per block (256 threads, wave32)
#define NCH    4    // channels

static __device__ __forceinline__ v8f wmma_f32(v2f a, v2f b, v8f c) {
    return __builtin_amdgcn_wmma_f32_16x16x4_f32(false, a, false, b,
                                                 (short)0, c, false, false);
}

__global__ __launch_bounds__(256) void pim_stage1(
    const float* __restrict__ pred, const float* __restrict__ tgt,
    float* __restrict__ partial, int T, int gpw, int rem)
{
    __shared__ float red[WPB][24];

    const int tid  = threadIdx.x;
    const int lane = tid & 31;
    // Force the wave index scalar so all loop control lives in SALU and the
    // WMMA runs under full EXEC with no per-iteration vector compares.
    const int wave = __builtin_amdgcn_readfirstlane(tid >> 5);
    const int b    = blockIdx.y;
    const int sx   = blockIdx.x;

    // Per-lane decode of the 16x16x4 f32 A/B operand layout:
    //   A (16x4): lanes 0-15 hold K=0,1 in v0,v1 ; lanes 16-31 hold K=2,3.
    //   B (4x16): lanes 0-15 hold K=0,1 ; lanes 16-31 hold K=2,3 (N = lane&15).
    const int kb = (lane >> 4) * 2;   // K base for this lane's 2 VGPRs
    const int mn = lane & 15;         // M (for A) == N (for B)
    const int a  = mn >> 3;           // K-chunk within the 8-sample group
    const int r  = mn & 7;
    const int ch = r & 3;
    const int hi = r >> 2;            // 0: A=pred,B=tgt ; 1: A=tgt,B=pred

    const size_t chanOff = ((size_t)(b * NCH + ch)) * (size_t)T
                         + (size_t)(4 * a + kb);
    const float* baseA = (hi ? tgt : pred) + chanOff;
    const float* baseB = (hi ? pred : tgt) + chanOff;

    // Scalar work distribution: wave w (within batch) gets a contiguous run of
    // gcnt 8-sample groups; first `rem` waves take one extra group.
    const int wb    = sx * WPB + wave;                       // scalar
    const int extra = (wb < rem) ? 1 : 0;                    // scalar
    const int gbeg  = wb * gpw + (wb < rem ? wb : rem);      // scalar
    const int gcnt  = gpw + extra;                           // scalar
    const int pairs = gcnt >> 1;                             // scalar

    const float* pa = baseA + (size_t)gbeg * 8;
    const float* pb = baseB + (size_t)gbeg * 8;

    v8f acc0 = {0.f, 0.f, 0.f, 0.f, 0.f, 0.f, 0.f, 0.f};
    v8f acc1 = acc0;

    if (pairs > 0) {
        // Software-pipelined: next iteration's 4 b64 loads issue before the
        // current pair of WMMAs, so s_wait only covers the older load group.
        v2f a0 = *(const v2f*)(pa),     b0 = *(const v2f*)(pb);
        v2f a1 = *(const v2f*)(pa + 8), b1 = *(const v2f*)(pb + 8);
        for (int i = 1; i < pairs; ++i) {
            pa += 16; pb += 16;
            v2f na0 = *(const v2f*)(pa),     nb0 = *(const v2f*)(pb);
            v2f na1 = *(const v2f*)(pa + 8), nb1 = *(const v2f*)(pb + 8);
            acc0 = wmma_f32(a0, b0, acc0);
            acc1 = wmma_f32(a1, b1, acc1);
            a0 = na0; b0 = nb0; a1 = na1; b1 = nb1;
        }
        acc0 = wmma_f32(a0, b0, acc0);
        acc1 = wmma_f32(a1, b1, acc1);
        pa += 16; pb += 16;
    }
    if (gcnt & 1) {
        v2f a0 = *(const v2f*)(pa), b0 = *(const v2f*)(pb);
        acc0 = wmma_f32(a0, b0, acc0);
    }
    v8f acc = acc0 + acc1;

    // D layout: VGPR g holds row g (lanes 0-15) / row g+8 (lanes 16-31),
    // col = lane&15. Every useful element's chunk-a=1 twin lives at lane+24
    // in the same VGPR -> deterministic one-shuffle merge (owners = lanes 0-7).
    const int partner = (lane & 7) + 24;
    float m[8];
#pragma unroll
    for (int g2 = 0; g2 < 8; ++g2)
        m[g2] = acc[g2] + __shfl(acc[g2], partner, 32);

    // Slot map: 0..15 = X[i][j], 16..19 = Sp[i], 20..23 = St[j]
    if (lane < 4) {
#pragma unroll
        for (int i2 = 0; i2 < 4; ++i2)
            red[wave][i2 * 4 + lane] = m[i2];       // X[i][lane]
        red[wave][20 + lane] = m[4 + lane];         // St[lane] (row 4+j, col j)
    } else if (lane < 8) {
        red[wave][16 + (lane - 4)] = m[lane - 4];   // Sp[i]    (row i, col 4+i)
    }
    __syncthreads();

    if (tid < 24) {
        float s = 0.f;
#pragma unroll
        for (int w = 0; w < WPB; ++w) s += red[w][tid];
        partial[((size_t)b * NSLICE + sx) * 24 + tid] = s;
    }
}

__global__ void pim_stage2(const float* __restrict__ partial,
                           float* __restrict__ out, int T)
{
    __shared__ float s24[24];
    const int b = blockIdx.x, tid = threadIdx.x;

    if (tid < 24) {
        float s = 0.f;
        for (int sl = 0; sl < NSLICE; ++sl)
            s += partial[((size_t)b * NSLICE + sl) * 24 + tid];
        s24[tid] = s;
    }
    __syncthreads();

    if (tid == 0) {
        const float invT = 1.0f / (float)T;
        float cost[4][4];
#pragma unroll
        for (int i = 0; i < 4; ++i)
#pragma unroll
            for (int j = 0; j < 4; ++j)
                cost[i][j] = (s24[16 + i] + s24[20 + j]
                              - 2.0f * s24[i * 4 + j]) * invT;

        const int perms[24][4] = {
            {0,1,2,3},{0,1,3,2},{0,2,1,3},{0,2,3,1},{0,3,1,2},{0,3,2,1},
            {1,0,2,3},{1,0,3,2},{1,2,0,3},{1,2,3,0},{1,3,0,2},{1,3,2,0},
            {2,0,1,3},{2,0,3,1},{2,1,0,3},{2,1,3,0},{2,3,0,1},{2,3,1,0},
            {3,0,1,2},{3,0,2,1},{3,1,0,2},{3,1,2,0},{3,2,0,1},{3,2,1,0}};

        float best = 3.402823466e38f;
#pragma unroll
        for (int p = 0; p < 24; ++p) {
            float c = 0.25f * (cost[0][perms[p][0]] + cost[1][perms[p][1]] +
                               cost[2][perms[p][2]] + cost[3][perms[p][3]]);
            best = fminf(best, c);
        }
        out[b] = best;
    }
}

extern "C" void kernel_launch(void* const* d_in, const int* in_sizes, int n_in,
                              void* d_out, int out_size, void* d_ws, size_t ws_size,
                              hipStream_t stream) {
    const float* pred = (const float*)d_in[0];
    const float* tgt  = (const float*)d_in[1];
    float* out        = (float*)d_out;
    float* partial    = (float*)d_ws;          // B*NSLICE*24 floats = 48 KB

    const int B = out_size;                    // 16
    const int T = in_sizes[0] / (B * NCH);     // 256000
    const int G = T / 8;                       // 8-sample groups (T % 8 == 0)
    const int wavesPerBatch = NSLICE * WPB;    // 256
    const int gpw = G / wavesPerBatch;         // 125
    const int rem = G % wavesPerBatch;         // 0 for T=256000

    pim_stage1<<<dim3(NSLICE, B), WPB * 32, 0, stream>>>(pred, tgt, partial, T, gpw, rem);
    pim_stage2<<<B, 32, 0, stream>>>(partial, out, T);
}